// Conv2d_58763742544721
// MI455X (gfx1250) — compile-verified
//
#include <hip/hip_runtime.h>

typedef float v2f __attribute__((ext_vector_type(2)));
typedef float v8f __attribute__((ext_vector_type(8)));

#define B_    32
#define CIN   32
#define H_    128
#define W_    128
#define COUT  64
#define HALF_C 8                       // channels per LDS slab (4 interleaved pairs)
#define N_HALF (CIN / HALF_C)          // 4 slabs
#define NPAIR  (HALF_C / 2)            // 4 channel-pairs per slab
#define ROWS   4                       // staged input rows: h-1 .. h+2
#define IN_P_STRIDE2 528               // float2 stride per channel-pair; 2*528 % 64 == 32 -> bank-clean
#define W_P_STRIDE2  80                // float2 stride per pair of weight rows; 2*80 % 64 == 32
#define IN_ELEMS2 (NPAIR * ROWS * 130) // 2080 float2 staged per slab
#define W_ELEMS2  (9 * NPAIR * COUT)   // 2304 float2 staged per slab

__global__ __launch_bounds__(256)
void conv3x3_wmma_f32(const float* __restrict__ x,
                      const float* __restrict__ wgt,
                      const float* __restrict__ bias,
                      float* __restrict__ out)
{
    __shared__ v2f sIn2[NPAIR * IN_P_STRIDE2];      // [pair][(row)*130 + xx] interleaved ch pair
    __shared__ v2f sW2 [9 * NPAIR * W_P_STRIDE2];   // [khw][pair][cout]      interleaved ch pair

    const int tid   = threadIdx.x;
    const int lane  = tid & 31;
    const int wave  = tid >> 5;
    const int nLo   = lane & 15;      // M / N index within half-wave
    const int hiSel = lane >> 4;      // 0: lanes 0-15, 1: lanes 16-31
    const int hrow  = wave >> 2;      // which of the block's two output rows
    const int x0    = (wave & 3) * 32;// this wave's 32-pixel span (two 16-px N tiles)

    const int b = blockIdx.x >> 6;    // blockIdx = b*64 + hpair
    const int h = (blockIdx.x & 63) * 2;

    v8f acc[8] = {v8f{}, v8f{}, v8f{}, v8f{}, v8f{}, v8f{}, v8f{}, v8f{}};

    for (int half = 0; half < N_HALF; ++half) {
        const int cg0 = half * HALF_C;

        // ---- stage input slab, adjacent channels interleaved into float2 ----
        for (int idx = tid; idx < IN_ELEMS2; idx += 256) {
            int p   = idx / 520;            // channel pair
            int rem = idx - p * 520;
            int r   = rem / 130;            // 0..3 -> rows h-1..h+2
            int xx  = rem - r * 130;        // 0..129 -> w-1..w+128
            int hh  = h - 1 + r;
            int ww  = xx - 1;
            v2f v = {0.0f, 0.0f};
            if ((unsigned)hh < H_ && (unsigned)ww < W_) {
                size_t base = (((size_t)b * CIN + (cg0 + 2 * p)) * H_ + hh) * W_ + ww;
                v.x = x[base];
                v.y = x[base + (size_t)H_ * W_];   // next channel
            }
            sIn2[p * IN_P_STRIDE2 + r * 130 + xx] = v;
        }
        // ---- stage weights as [khw][pair][cout] float2 ----
        for (int idx = tid; idx < W_ELEMS2; idx += 256) {
            int khw = idx >> 8;             // / 256
            int rem = idx & 255;
            int p   = rem >> 6;             // / 64
            int co  = rem & 63;
            size_t gb = ((size_t)co * CIN + (cg0 + 2 * p)) * 9 + khw;
            v2f v;
            v.x = wgt[gb];
            v.y = wgt[gb + 9];              // next channel, same (co,khw)
            sW2[khw * (NPAIR * W_P_STRIDE2) + p * W_P_STRIDE2 + co] = v;
        }
        // ---- prefetch next slab's input while computing this one ----
        if (half + 1 < N_HALF) {
            for (int idx = tid * 16; idx < IN_ELEMS2; idx += 256 * 16) {
                int p   = idx / 520;
                int rem = idx - p * 520;
                int r   = rem / 130;
                int xx  = rem - r * 130;
                int hh  = h - 1 + r;
                int ww  = xx - 1;
                if ((unsigned)hh < H_ && (unsigned)ww < W_)
                    __builtin_prefetch(
                        &x[(((size_t)b * CIN + (cg0 + HALF_C + 2 * p)) * H_ + hh) * W_ + ww],
                        0, 1);
            }
        }
        __syncthreads();

        // ---- compute: K = (kh,kw) x ci, stepped by 4 for 16x16x4 f32 WMMA ----
        #pragma unroll
        for (int kh = 0; kh < 3; ++kh) {
            #pragma unroll
            for (int kw = 0; kw < 3; ++kw) {
                const int khw = kh * 3 + kw;
                const v2f* wB = &sW2[khw * (NPAIR * W_P_STRIDE2)];
                const v2f* iB = &sIn2[(kh + hrow) * 130 + x0 + nLo + kw];
                #pragma unroll
                for (int c4 = 0; c4 < HALF_C; c4 += 4) {
                    const int p = (c4 >> 1) + hiSel;   // K pair: .x=K, .y=K+1
                    // B fragments: one aligned b64 load each, two N tiles
                    v2f bf0 = iB[p * IN_P_STRIDE2];
                    v2f bf1 = iB[p * IN_P_STRIDE2 + 16];
                    // A fragments: one aligned b64 load per 16-row M tile
                    const v2f* wR = &wB[p * W_P_STRIDE2 + nLo];
                    v2f a0 = wR[ 0];
                    v2f a1 = wR[16];
                    v2f a2 = wR[32];
                    v2f a3 = wR[48];
                    acc[0] = __builtin_amdgcn_wmma_f32_16x16x4_f32(false, a0, false, bf0, (short)0, acc[0], false, false);
                    acc[1] = __builtin_amdgcn_wmma_f32_16x16x4_f32(false, a1, false, bf0, (short)0, acc[1], false, false);
                    acc[2] = __builtin_amdgcn_wmma_f32_16x16x4_f32(false, a2, false, bf0, (short)0, acc[2], false, false);
                    acc[3] = __builtin_amdgcn_wmma_f32_16x16x4_f32(false, a3, false, bf0, (short)0, acc[3], false, false);
                    acc[4] = __builtin_amdgcn_wmma_f32_16x16x4_f32(false, a0, false, bf1, (short)0, acc[4], false, false);
                    acc[5] = __builtin_amdgcn_wmma_f32_16x16x4_f32(false, a1, false, bf1, (short)0, acc[5], false, false);
                    acc[6] = __builtin_amdgcn_wmma_f32_16x16x4_f32(false, a2, false, bf1, (short)0, acc[6], false, false);
                    acc[7] = __builtin_amdgcn_wmma_f32_16x16x4_f32(false, a3, false, bf1, (short)0, acc[7], false, false);
                }
            }
        }
        __syncthreads();
    }

    // ---- epilogue: cout = m*16 + r + 8*hiSel; pixels x0+nLo and x0+16+nLo ----
    const int h_out = h + hrow;
    const size_t rowBase = ((size_t)b * COUT) * (H_ * W_) + (size_t)h_out * W_;
    #pragma unroll
    for (int m = 0; m < 4; ++m) {
        #pragma unroll
        for (int r = 0; r < 8; ++r) {
            const int co = m * 16 + r + 8 * hiSel;
            const float bv = bias[co];
            const size_t o = rowBase + (size_t)co * (H_ * W_) + x0 + nLo;
            out[o]      = acc[m][r]     + bv;
            out[o + 16] = acc[4 + m][r] + bv;
        }
    }
}

extern "C" void kernel_launch(void* const* d_in, const int* in_sizes, int n_in,
                              void* d_out, int out_size, void* d_ws, size_t ws_size,
                              hipStream_t stream)
{
    const float* x    = (const float*)d_in[0];  // [32,32,128,128]
    const float* wgt  = (const float*)d_in[1];  // [64,32,3,3]
    const float* bias = (const float*)d_in[2];  // [64]
    float* out        = (float*)d_out;          // [32,64,128,128]

    dim3 grid(B_ * H_ / 2);  // 2048 workgroups: one per (b, row-pair)
    dim3 block(256);         // 8 wave32 waves, each owning 32 pixels x all 64 cout
    conv3x3_wmma_f32<<<grid, block, 0, stream>>>(x, wgt, bias, out);
}